// CustomPrecision_78907139162809
// MI455X (gfx1250) — compile-verified
//
#include <hip/hip_runtime.h>
#include <hip/hip_bf16.h>

#define NUM_CLASSES 32
#define HBINS (NUM_CLASSES * NUM_CLASSES)

typedef __attribute__((ext_vector_type(2))) float v2f;
typedef __attribute__((ext_vector_type(8))) float v8f;

// ---------------------------------------------------------------------------
// Kernel 0: zero the global histogram in workspace (ws is poisoned 0xAA).
// Integer counters => replay-deterministic regardless of atomic ordering.
// ---------------------------------------------------------------------------
__global__ void cm_zero_kernel(unsigned int* __restrict__ hist) {
    int i = blockIdx.x * blockDim.x + threadIdx.x;
    if (i < HBINS) hist[i] = 0u;
}

// ---------------------------------------------------------------------------
// First-occurrence argmax over a 128-byte row using float4 (b128) loads.
// Strict '>' reproduces jnp.argmax tie-breaking (first max wins).
// ---------------------------------------------------------------------------
__device__ __forceinline__ int argmax32(const float* __restrict__ row) {
    const float4* r4 = (const float4*)row;
    float best = -__builtin_inff();
    int bi = 0;
#pragma unroll
    for (int q = 0; q < 8; ++q) {
        float4 v = r4[q];
        int b = q * 4;
        if (v.x > best) { best = v.x; bi = b + 0; }
        if (v.y > best) { best = v.y; bi = b + 1; }
        if (v.z > best) { best = v.z; bi = b + 2; }
        if (v.w > best) { best = v.w; bi = b + 3; }
    }
    return bi;
}

// ---------------------------------------------------------------------------
// Kernel 1: bandwidth-bound streamer. 512 MB of reads -> ~22 us at 23.3 TB/s.
// LDS histogram per block (ds_add atomics), one global integer-atomic flush
// per block at the end. global_prefetch_b8 one grid-stride ahead.
// ---------------------------------------------------------------------------
__global__ void cm_hist_kernel(const float* __restrict__ y_true,
                               const float* __restrict__ y_pred,
                               unsigned int* __restrict__ ghist,
                               int n_rows) {
    __shared__ unsigned int lhist[HBINS];
    for (int i = threadIdx.x; i < HBINS; i += blockDim.x) lhist[i] = 0u;
    __syncthreads();

    const long long stride = (long long)gridDim.x * blockDim.x;
    for (long long r = (long long)blockIdx.x * blockDim.x + threadIdx.x;
         r < n_rows; r += stride) {
        long long nr = r + stride;
        if (nr < n_rows) {
            __builtin_prefetch(y_true + nr * NUM_CLASSES, 0, 0);
            __builtin_prefetch(y_pred + nr * NUM_CLASSES, 0, 0);
        }
        int t = argmax32(y_true + r * NUM_CLASSES);
        int p = argmax32(y_pred + r * NUM_CLASSES);
        atomicAdd(&lhist[t * NUM_CLASSES + p], 1u);
    }

    __syncthreads();
    for (int i = threadIdx.x; i < HBINS; i += blockDim.x) {
        unsigned int c = lhist[i];
        if (c) atomicAdd(&ghist[i], c);
    }
}

// ---------------------------------------------------------------------------
// Kernel 2: single-wave finalize. Column sums of the 32x32 cm via
// V_WMMA_F32_16X16X4_F32 with A = ones(16x4):  D[m][n] = sum_k B[k][n] + C.
// Accumulate 8 K-chunks (4 rows each) for two 16-column blocks; every row of
// D carries the colsums, so lane L reads its own colsum from element 0.
// Exact f32 math (counts < 2^24) => bit-exact colsums.
//
// VGPR layouts (wave32, ISA 7.12.2):
//   A 16x4 f32 : VGPR0 = K0 (lanes 0-15) / K2 (lanes 16-31); VGPR1 = K1/K3.
//                All-ones => a = {1,1}.
//   B 4x16 f32 : VGPR0 = row K0 (lanes 0-15) / K2 (16-31); VGPR1 = K1/K3,
//                N = lane & 15.
//   D 16x16 f32: VGPR0 lanes 0-15 = (M=0, N=lane); lanes 16-31 = (M=8, N-16).
// ---------------------------------------------------------------------------
__global__ void cm_finalize_kernel(const unsigned int* __restrict__ hist,
                                   const float* __restrict__ conf,
                                   float* __restrict__ out) {
    const int lane = threadIdx.x;      // 0..31, exactly one wave
    const int n    = lane & 15;
    const int half = lane >> 4;

    __shared__ float cm[HBINS];
    for (int i = lane; i < HBINS; i += 32)
        cm[i] = (float)hist[i] + conf[i];
    __syncthreads();

    v2f a;
    a[0] = 1.0f;
    a[1] = 1.0f;
    v8f acc0 = {};   // columns 0..15
    v8f acc1 = {};   // columns 16..31

#pragma unroll
    for (int kc = 0; kc < 8; ++kc) {
        const int r0 = kc * 4 + half * 2;     // row for B VGPR0 in this lane-half
        v2f b0, b1;
        b0[0] = cm[(r0 + 0) * NUM_CLASSES + n];
        b0[1] = cm[(r0 + 1) * NUM_CLASSES + n];
        b1[0] = cm[(r0 + 0) * NUM_CLASSES + 16 + n];
        b1[1] = cm[(r0 + 1) * NUM_CLASSES + 16 + n];
        acc0 = __builtin_amdgcn_wmma_f32_16x16x4_f32(
            false, a, false, b0, (short)0, acc0, false, false);
        acc1 = __builtin_amdgcn_wmma_f32_16x16x4_f32(
            false, a, false, b1, (short)0, acc1, false, false);
    }

    // Lane L<16: acc0[0] = colsum[L]; lane L>=16: acc1[0] = colsum[L].
    const float colsum = (half == 0) ? acc0[0] : acc1[0];
    const float tp = cm[lane * NUM_CLASSES + lane];
    const float eps = 1.1920928955078125e-07f;   // np.finfo(float32).eps
    float prec = tp / (colsum + eps);

    // wave32 reduction for the mean
#pragma unroll
    for (int m = 16; m >= 1; m >>= 1)
        prec += __shfl_xor(prec, m, 32);

    if (lane == 0) out[0] = prec * (1.0f / 32.0f);
}

extern "C" void kernel_launch(void* const* d_in, const int* in_sizes, int n_in,
                              void* d_out, int out_size, void* d_ws, size_t ws_size,
                              hipStream_t stream) {
    const float* y_true = (const float*)d_in[0];
    const float* y_pred = (const float*)d_in[1];
    const float* conf   = (const float*)d_in[2];   // 32x32 f32 (zeros)
    float* out = (float*)d_out;
    unsigned int* hist = (unsigned int*)d_ws;

    const int n_rows = in_sizes[0] / NUM_CLASSES;

    cm_zero_kernel<<<(HBINS + 255) / 256, 256, 0, stream>>>(hist);
    cm_hist_kernel<<<1024, 256, 0, stream>>>(y_true, y_pred, hist, n_rows);
    cm_finalize_kernel<<<1, 32, 0, stream>>>(hist, conf, out);
}